// SurfaceConsistencyLoss_60739427500438
// MI455X (gfx1250) — compile-verified
//
#include <hip/hip_runtime.h>
#include <math.h>

typedef __attribute__((ext_vector_type(2))) float v2f;
typedef __attribute__((ext_vector_type(8))) float v8f;

#define BB 4
#define NN 50000
#define KK 128
#define GDIM 128
#define TILES ((NN + 15) / 16)          // 3125 (exact)
#define WPB 8                           // waves per block (256 threads)

// Use gfx1250 async global->LDS copy (ASYNCcnt-tracked, no VGPR round trip).
#define USE_ASYNC_LDS 1

__device__ __forceinline__ void async_ld_b128(unsigned lds_addr, unsigned byte_off,
                                              const void* gbase) {
  // GVS mode: mem_addr = SADDR(64) + VADDR(32) + IOFFSET ; VDST = LDS byte addr
  asm volatile("global_load_async_to_lds_b128 %0, %1, %2"
               :
               : "v"(lds_addr), "v"(byte_off), "s"(gbase)
               : "memory");
}

__device__ __forceinline__ void wait_asynccnt0() {
#if __has_builtin(__builtin_amdgcn_s_wait_asynccnt)
  __builtin_amdgcn_s_wait_asynccnt(0);
#else
  asm volatile("s_wait_asynccnt 0x0" ::: "memory");
#endif
}

// ---------------------------------------------------------------------------
// Pass 1: per-(b,k) constants.
//   M_k = R_t R^T  (9),  v_k = c_t - M_k c  (3),  ones column, pad to 16.
//   kc  = (cx, cy, cz, 1/(2 s^2), const^2, pad...)
// ---------------------------------------------------------------------------
__global__ __launch_bounds__(256) void scl_precompute(
    const float* __restrict__ constants, const float* __restrict__ scales,
    const float* __restrict__ rotations, const float* __restrict__ centers,
    float* __restrict__ Vmat, float* __restrict__ kcst) {
  int t = blockIdx.x * blockDim.x + threadIdx.x;
  if (t >= BB * KK) return;
  int b = t / KK, k = t % KK;
  int bt = (b + 1) % BB;

  float R[9], Rt[9];
#pragma unroll
  for (int i = 0; i < 9; ++i) {
    R[i]  = rotations[(size_t)(b  * KK + k) * 9 + i];
    Rt[i] = rotations[(size_t)(bt * KK + k) * 9 + i];
  }
  // R_rel[i][l] = sum_j Rt[i][j] * R[l][j]
  float M[9];
#pragma unroll
  for (int i = 0; i < 3; ++i)
#pragma unroll
    for (int l = 0; l < 3; ++l)
      M[i * 3 + l] = Rt[i * 3 + 0] * R[l * 3 + 0] +
                     Rt[i * 3 + 1] * R[l * 3 + 1] +
                     Rt[i * 3 + 2] * R[l * 3 + 2];

  float c0  = centers[(size_t)(b  * KK + k) * 3 + 0];
  float c1  = centers[(size_t)(b  * KK + k) * 3 + 1];
  float c2  = centers[(size_t)(b  * KK + k) * 3 + 2];
  float ct0 = centers[(size_t)(bt * KK + k) * 3 + 0];
  float ct1 = centers[(size_t)(bt * KK + k) * 3 + 1];
  float ct2 = centers[(size_t)(bt * KK + k) * 3 + 2];

  float v0 = ct0 - (M[0] * c0 + M[1] * c1 + M[2] * c2);
  float v1 = ct1 - (M[3] * c0 + M[4] * c1 + M[5] * c2);
  float v2 = ct2 - (M[6] * c0 + M[7] * c1 + M[8] * c2);

  float* vr = Vmat + (size_t)(b * KK + k) * 16;
#pragma unroll
  for (int i = 0; i < 9; ++i) vr[i] = M[i];
  vr[9]  = v0; vr[10] = v1; vr[11] = v2;
  vr[12] = 1.0f;                       // ones column -> D[:,12] = sum_k w
  vr[13] = 0.0f; vr[14] = 0.0f; vr[15] = 0.0f;

  float s  = scales[(size_t)(b * KK + k)];
  float cc = constants[(size_t)(b * KK + k)];
  float* kr = kcst + (size_t)(b * KK + k) * 8;
  kr[0] = c0; kr[1] = c1; kr[2] = c2;
  kr[3] = 1.0f / (2.0f * s * s);
  kr[4] = cc * cc;
  kr[5] = 0.0f; kr[6] = 0.0f; kr[7] = 0.0f;
}

// ---------------------------------------------------------------------------
// Pass 2: per 16-point tile, W(16x128) x V(128x16) via 32x V_WMMA_F32_16X16X4_F32
// then trilinear sample epilogue + loss reduction.
// ---------------------------------------------------------------------------
__global__ __launch_bounds__(256) void scl_main(
    const float* __restrict__ samples,   // (B,N,6)
    const float* __restrict__ grid,      // (B,GD,GD,GD)
    const float* __restrict__ w2g,       // (B,4,4)
    const float* __restrict__ Vmat,      // (B,K,16)
    const float* __restrict__ kcst,      // (B,K,8)
    float* __restrict__ out) {
  __shared__ float sV[KK * 16];          // 8 KB
  __shared__ float sK[KK * 8];           // 4 KB
  __shared__ float sD[WPB][16][17];      // D transpose staging (padded)

  const int b   = blockIdx.y;
  const int tid = threadIdx.x;

#if USE_ASYNC_LDS
  {
    // Stage k-constant tables with async global->LDS B128 copies (ASYNCcnt).
    // sV: 8 KB = 512 x b128 ; sK: 4 KB = 256 x b128.
    const unsigned ldsV = (unsigned)(size_t)(&sV[0]);
    const unsigned ldsK = (unsigned)(size_t)(&sK[0]);
    const void* gV = (const void*)(Vmat + (size_t)b * KK * 16);
    const void* gK = (const void*)(kcst + (size_t)b * KK * 8);
#pragma unroll
    for (int c = 0; c < 2; ++c) {
      unsigned off = (unsigned)(tid + c * 256) * 16u;
      async_ld_b128(ldsV + off, off, gV);
    }
    {
      unsigned off = (unsigned)tid * 16u;
      if (tid < 256) async_ld_b128(ldsK + off, off, gK);
    }
    wait_asynccnt0();
  }
#else
  for (int i = tid; i < KK * 16; i += 256) sV[i] = Vmat[(size_t)b * KK * 16 + i];
  for (int i = tid; i < KK * 8;  i += 256) sK[i] = kcst[(size_t)b * KK * 8 + i];
#endif
  __syncthreads();

  const int wave = tid >> 5;
  const int lane = tid & 31;
  const int tile = blockIdx.x * WPB + wave;
  const bool active = (tile < TILES);    // wave-uniform: EXEC stays all-1 for WMMA

  if (active) {
    const int m = lane & 15;
    int n = tile * 16 + m; if (n >= NN) n = NN - 1;
    const float* sp = samples + ((size_t)b * NN + n) * 6;
    const float px = sp[0], py = sp[1], pz = sp[2];

    const int khalf = (lane >> 4) * 2;   // A/B 16x16x4 f32 lane layout: K 0/1 vs 2/3
    const int col   = lane & 15;
    v8f acc = {};

    for (int k0 = 0; k0 < KK; k0 += 4) {
      const int ka = k0 + khalf;
      const float* kr0 = sK + (size_t)ka * 8;
      const float* kr1 = sK + (size_t)(ka + 1) * 8;
      float dx = px - kr0[0], dy = py - kr0[1], dz = pz - kr0[2];
      float w0 = __expf(-(dx * dx + dy * dy + dz * dz) * kr0[3]) * kr0[4];
      dx = px - kr1[0]; dy = py - kr1[1]; dz = pz - kr1[2];
      float w1 = __expf(-(dx * dx + dy * dy + dz * dz) * kr1[3]) * kr1[4];

      v2f a; a.x = w0; a.y = w1;                       // A[m][ka], A[m][ka+1]
      v2f bf; bf.x = sV[ka * 16 + col];                // B[ka][col]
      bf.y = sV[(ka + 1) * 16 + col];                  // B[ka+1][col]

      acc = __builtin_amdgcn_wmma_f32_16x16x4_f32(
          /*neg_a=*/false, a, /*neg_b=*/false, bf,
          /*c_mod=*/(short)0, acc, /*reuse_a=*/false, /*reuse_b=*/false);
    }

    // D layout: VGPR r holds row r (lanes 0-15) / row r+8 (lanes 16-31), col = lane&15
    const int rbase = (lane >= 16) ? 8 : 0;
#pragma unroll
    for (int r = 0; r < 8; ++r) sD[wave][rbase + r][lane & 15] = acc[r];
  }
  __syncthreads();

  float val = 0.0f;
  if (active && lane < 16) {
    const int m = lane;
    const int n = tile * 16 + m;
    if (n < NN) {
      const float* Dr = &sD[wave][m][0];
      const float invS = 1.0f / Dr[12];
      const float* sp = samples + ((size_t)b * NN + n) * 6;
      const float p0 = sp[0], p1 = sp[1], p2 = sp[2];
      float bl[3];
#pragma unroll
      for (int i = 0; i < 3; ++i)
        bl[i] = (Dr[3 * i + 0] * p0 + Dr[3 * i + 1] * p1 + Dr[3 * i + 2] * p2 +
                 Dr[9 + i]) * invS;

      const int bt = (b + 1) & (BB - 1);
      const float* w = w2g + (size_t)bt * 16;
      float g[3];
#pragma unroll
      for (int i = 0; i < 3; ++i)
        g[i] = w[i * 4 + 0] * bl[0] + w[i * 4 + 1] * bl[1] +
               w[i * 4 + 2] * bl[2] + w[i * 4 + 3];

      const float denom = (float)(GDIM - 1);
      // normalize to [-1,1] then unnormalize with border clamp (as reference)
      float gx = 2.0f * (g[0] / denom) - 1.0f;
      float gy = 2.0f * (g[1] / denom) - 1.0f;
      float gz = 2.0f * (g[2] / denom) - 1.0f;
      float x = fminf(fmaxf((gx + 1.0f) * 0.5f * denom, 0.0f), denom);
      float y = fminf(fmaxf((gy + 1.0f) * 0.5f * denom, 0.0f), denom);
      float z = fminf(fmaxf((gz + 1.0f) * 0.5f * denom, 0.0f), denom);

      float x0 = floorf(x), y0 = floorf(y), z0 = floorf(z);
      float fx = x - x0, fy = y - y0, fz = z - z0;
      int x0i = (int)x0, y0i = (int)y0, z0i = (int)z0;
      int x1i = min(x0i + 1, GDIM - 1);
      int y1i = min(y0i + 1, GDIM - 1);
      int z1i = min(z0i + 1, GDIM - 1);

      const float* vol = grid + (size_t)bt * GDIM * GDIM * GDIM;
#define VAT(zz, yy, xx) vol[(((size_t)(zz) * GDIM + (yy)) * GDIM + (xx))]
      float c000 = VAT(z0i, y0i, x0i), c001 = VAT(z0i, y0i, x1i);
      float c010 = VAT(z0i, y1i, x0i), c011 = VAT(z0i, y1i, x1i);
      float c100 = VAT(z1i, y0i, x0i), c101 = VAT(z1i, y0i, x1i);
      float c110 = VAT(z1i, y1i, x0i), c111 = VAT(z1i, y1i, x1i);
#undef VAT
      float c00 = c000 * (1.0f - fx) + c001 * fx;
      float c01 = c010 * (1.0f - fx) + c011 * fx;
      float c10 = c100 * (1.0f - fx) + c101 * fx;
      float c11 = c110 * (1.0f - fx) + c111 * fx;
      float c0 = c00 * (1.0f - fy) + c01 * fy;
      float c1 = c10 * (1.0f - fy) + c11 * fy;
      float sdf = c0 * (1.0f - fz) + c1 * fz;
      val = sdf * sdf * (1.0f / (float)NN);
    }
  }

  // wave32 shuffle reduction, one atomic per wave
#pragma unroll
  for (int off = 16; off > 0; off >>= 1) val += __shfl_down(val, off, 32);
  if (lane == 0 && active) atomicAdd(out, val);
}

// ---------------------------------------------------------------------------
extern "C" void kernel_launch(void* const* d_in, const int* in_sizes, int n_in,
                              void* d_out, int out_size, void* d_ws, size_t ws_size,
                              hipStream_t stream) {
  const float* constants = (const float*)d_in[0];   // (B,K)
  const float* scales    = (const float*)d_in[1];   // (B,K)
  const float* rotations = (const float*)d_in[2];   // (B,K,3,3)
  const float* centers   = (const float*)d_in[3];   // (B,K,3)
  const float* samples   = (const float*)d_in[4];   // (B,N,6)
  const float* grid      = (const float*)d_in[5];   // (B,GD,GD,GD)
  const float* w2g       = (const float*)d_in[6];   // (B,4,4)
  float* out = (float*)d_out;

  float* wsf  = (float*)d_ws;
  float* Vmat = wsf;                        // B*K*16 floats
  float* kcst = wsf + (size_t)BB * KK * 16; // B*K*8 floats

  hipMemsetAsync(out, 0, sizeof(float), stream);

  scl_precompute<<<dim3((BB * KK + 255) / 256), dim3(256), 0, stream>>>(
      constants, scales, rotations, centers, Vmat, kcst);

  dim3 g((TILES + WPB - 1) / WPB, BB);
  scl_main<<<g, dim3(256), 0, stream>>>(samples, grid, w2g, Vmat, kcst, out);
}